// tt_LinearV2_54254026883465
// MI455X (gfx1250) — compile-verified
//
#include <hip/hip_runtime.h>
#include <hip/hip_bf16.h>

// TT-linear forward, fully fused per token in LDS, fp32 WMMA (16x16x4) on gfx1250.
//
// Per token (one workgroup, 8 wave32s):
//   stage X[4096] + all cores into LDS (async global->LDS, ASYNCcnt-tracked)
//   T1[b,c,d,e][i,v]   = X        x core0   (M=1024,K= 4,N=32)
//   T2[c,d,e][v][j,w]  = T1       x core1   (M=1024,K=32,N=32)
//   T3[d,e][v,w][k,x]  = T2       x core2   (M= 512,K=64,N=64)
//   T4[e][v,w,x][l,y]  = T3       x core3   (M= 512,K=64,N=64)
//   out[v,w,x,y][z]    = T4       x core4p  (M=1024,K=32,N=16, z<4 valid) + bias

typedef float v2f __attribute__((ext_vector_type(2)));
typedef float v8f __attribute__((ext_vector_type(8)));

#define NWAVES 8

#if defined(__has_builtin)
#if __has_builtin(__builtin_amdgcn_global_load_async_to_lds_b128)
#define HAVE_ASYNC_LDS 1
#endif
#if __has_builtin(__builtin_amdgcn_s_wait_asynccnt)
#define HAVE_WAIT_ASYNC_BUILTIN 1
#endif
#endif

#ifdef HAVE_ASYNC_LDS
#define AS1 __attribute__((address_space(1)))
#define AS3 __attribute__((address_space(3)))
typedef int v4i_b128 __attribute__((vector_size(16)));  // matches builtin pointee type
// 16B per lane, global -> LDS, no VGPR round trip; tracked by ASYNCcnt.
__device__ __forceinline__ void async_copy16(const float* g, float* l) {
  __builtin_amdgcn_global_load_async_to_lds_b128(
      (AS1 v4i_b128*)const_cast<float*>(g), (AS3 v4i_b128*)l,
      /*offset=*/0, /*cpol=*/0);
}
__device__ __forceinline__ void wait_async0() {
#ifdef HAVE_WAIT_ASYNC_BUILTIN
  __builtin_amdgcn_s_wait_asynccnt(0);
#else
  asm volatile("s_wait_asynccnt 0x0" ::: "memory");
#endif
}
#endif

// A-fragment address (element index into the step's LDS input tensor) for row M, col K.
__device__ __forceinline__ int a_addr(int STEP, int M, int K) {
  switch (STEP) {
    case 0:  // X flat [a][bcde]: a*1024 + bcde
      return (K << 10) + M;
    case 1:  // T1: b*8192 + cde*32 + i*4 + v ; M=(cde,v), K=(b,i)
      return ((K >> 3) << 13) + ((M >> 2) << 5) + ((K & 7) << 2) + (M & 3);
    case 2:  // T2: c*4096 + de*128 + v*32 + j*4 + w ; M=(de,v,w), K=(c,j)
      return ((K >> 3) << 12) + ((M >> 4) << 7) + (((M >> 2) & 3) << 5) + ((K & 7) << 2) + (M & 3);
    case 3:  // T3: d*4096 + e*1024 + v*256 + w*64 + k*8 + x ; M=(e,v,w,x), K=(d,k)
      return ((K >> 3) << 12) + ((M >> 7) << 10) + (((M >> 5) & 3) << 8) + (((M >> 3) & 3) << 6) + ((K & 7) << 3) + (M & 7);
    default: // T4: e*8192 + v*2048 + w*512 + x*64 + l*8 + y ; M=(v,w,x,y), K=(e,l)
      return ((K >> 3) << 13) + ((M >> 8) << 11) + (((M >> 6) & 3) << 9) + (((M >> 3) & 7) << 6) + ((K & 7) << 3) + (M & 7);
  }
}

// B-fragment address (element index into the step's LDS core) for row K, col N.
__device__ __forceinline__ int b_addr(int STEP, int K, int N) {
  switch (STEP) {
    case 0:  return (K << 5) + N;                                 // core0[a][i*4+v]
    case 1:  return ((K >> 3) << 8) + ((K & 7) << 5) + N;         // core1[b][i][j*4+w]
    case 2:
    case 3:  return ((K >> 3) << 9) + ((K & 7) << 6) + N;         // core2/3[c][j][k*8+x]
    default: return (K << 4) + N;                                 // core4 padded [K][16]
  }
}

template <int STEP, int MT, int NT, int KCH, int LDO>
__device__ __forceinline__ void tt_step(const float* __restrict__ A,
                                        const float* __restrict__ B,
                                        float* __restrict__ O,
                                        float* __restrict__ gout,
                                        const float* __restrict__ bias,
                                        int wave, int lane) {
  const int lm = lane & 15;            // column within 16-wide tile
  const int kk = (lane >> 4) << 1;     // K sub-offset for A/B fragments (0 or 2)
  const int rb = (lane >> 4) << 3;     // C/D row base offset (0 or 8)
  constexpr int WPN = NWAVES / NT;     // waves sharing one N-tile column
  const int nt   = wave / WPN;
  const int msub = wave % WPN;
  const int ncol = nt * 16 + lm;

  // Cache all B fragments for this wave's N-tile across the whole K range.
  v2f bfrag[KCH];
#pragma unroll
  for (int kc = 0; kc < KCH; ++kc) {
    const int k0 = kc * 4 + kk;
    bfrag[kc].x = B[b_addr(STEP, k0,     ncol)];
    bfrag[kc].y = B[b_addr(STEP, k0 + 1, ncol)];
  }

  for (int mt = msub; mt < MT; mt += WPN) {
    const int mrow = mt * 16 + lm;
    v8f acc = {};
#pragma unroll
    for (int kc = 0; kc < KCH; ++kc) {
      const int k0 = kc * 4 + kk;
      v2f afrag;
      afrag.x = A[a_addr(STEP, mrow, k0)];
      afrag.y = A[a_addr(STEP, mrow, k0 + 1)];
      acc = __builtin_amdgcn_wmma_f32_16x16x4_f32(false, afrag, false, bfrag[kc],
                                                  (short)0, acc, false, false);
    }
    const int rbase = mt * 16 + rb;
    if (STEP < 4) {
#pragma unroll
      for (int r = 0; r < 8; ++r)
        O[(rbase + r) * LDO + ncol] = acc[r];
    } else {
      if (lm < 4) {  // only z = 0..3 columns are real (padded N)
#pragma unroll
        for (int r = 0; r < 8; ++r) {
          const int oi = (rbase + r) * 4 + lm;
          gout[oi] = acc[r] + bias[oi];
        }
      }
    }
  }
}

__global__ __launch_bounds__(256) void tt_linear_fused_kernel(
    const float* __restrict__ X,
    const float* __restrict__ c0, const float* __restrict__ c1,
    const float* __restrict__ c2, const float* __restrict__ c3,
    const float* __restrict__ c4, const float* __restrict__ bias,
    float* __restrict__ out) {
  extern __shared__ float smem[];
  float* buf0 = smem;            // 32768 f32
  float* buf1 = smem + 32768;    // 32768 f32 (ping-pong)
  float* L0 = smem + 65536;      // 128
  float* L1 = L0 + 128;          // 1024
  float* L2 = L1 + 1024;         // 4096
  float* L3 = L2 + 4096;         // 4096
  float* L4 = L3 + 4096;         // 512 (core4 zero-padded to 32x16)

  const int tid  = threadIdx.x;
  const int wave = tid >> 5;
  const int lane = tid & 31;
  const long n = blockIdx.x;
  const float* Xn = X + n * 4096;

#ifdef HAVE_ASYNC_LDS
  // Async global->LDS staging: 16B/lane moves, no VGPR round trip (ASYNCcnt).
  {
    const int q = tid * 4;                       // 256 threads x 4 floats = 1024 f32/round
    for (int i = q; i < 4096; i += 1024) async_copy16(Xn + i, buf1 + i);  // X: 4 rounds
    for (int i = q; i < 4096; i += 1024) async_copy16(c2 + i, L2 + i);    // core2
    for (int i = q; i < 4096; i += 1024) async_copy16(c3 + i, L3 + i);    // core3
    if (q < 1024) async_copy16(c1 + q, L1 + q);                           // core1
    if (q < 128)  async_copy16(c0 + q, L0 + q);                           // core0
  }
#else
  for (int i = tid; i < 128;  i += 256) L0[i] = c0[i];
  for (int i = tid; i < 1024; i += 256) L1[i] = c1[i];
  for (int i = tid; i < 4096; i += 256) L2[i] = c2[i];
  for (int i = tid; i < 4096; i += 256) L3[i] = c3[i];
  for (int i = tid; i < 4096; i += 256) buf1[i] = Xn[i];
#endif
  // core4 needs zero-padding from 32x4 to 32x16: build synchronously.
  for (int i = tid; i < 512; i += 256) {
    const int K = i >> 4, N = i & 15;  // core4 (e,l,1,z): e*32 + l*4 + z
    L4[i] = (N < 4) ? c4[((K >> 3) << 5) + ((K & 7) << 2) + N] : 0.0f;
  }
#ifdef HAVE_ASYNC_LDS
  wait_async0();  // our own async transfers landed in LDS
#endif
  __syncthreads();

  tt_step<0, 64, 2, 1,  32>(buf1, L0, buf0, nullptr, nullptr, wave, lane);
  __syncthreads();
  tt_step<1, 64, 2, 8,  32>(buf0, L1, buf1, nullptr, nullptr, wave, lane);
  __syncthreads();
  tt_step<2, 32, 4, 16, 64>(buf1, L2, buf0, nullptr, nullptr, wave, lane);
  __syncthreads();
  tt_step<3, 32, 4, 16, 64>(buf0, L3, buf1, nullptr, nullptr, wave, lane);
  __syncthreads();
  tt_step<4, 64, 1, 8,   4>(buf1, L4, nullptr, out + n * 4096, bias, wave, lane);
}

extern "C" void kernel_launch(void* const* d_in, const int* in_sizes, int n_in,
                              void* d_out, int out_size, void* d_ws, size_t ws_size,
                              hipStream_t stream) {
  const float* X    = (const float*)d_in[0];
  const float* c0   = (const float*)d_in[1];
  const float* c1   = (const float*)d_in[2];
  const float* c2   = (const float*)d_in[3];
  const float* c3   = (const float*)d_in[4];
  const float* c4   = (const float*)d_in[5];
  const float* bias = (const float*)d_in[6];
  float* out = (float*)d_out;

  const int n_tokens = in_sizes[0] / 4096;
  const size_t shmem = (size_t)(2 * 32768 + 128 + 1024 + 4096 + 4096 + 512) * sizeof(float);

  tt_linear_fused_kernel<<<dim3(n_tokens), dim3(256), shmem, stream>>>(
      X, c0, c1, c2, c3, c4, bias, out);
}